// _spikeLayer_65197603553542
// MI455X (gfx1250) — compile-verified
//
#include <hip/hip_runtime.h>
#include <stdint.h>

// SLAYER spike scan: [P, T] with T=100 innermost, P = 16*64*32*32 pixels.
// Memory-bound (~839 MB traffic -> ~36us at 23.3 TB/s). Strategy: stage each
// block's contiguous 128-pixel tile through LDS with CDNA5 async global<->LDS
// ops for full coalescing, do the serial per-pixel scan out of LDS.

#define T_STEPS          100
#define CHUNKS_PER_PIXEL (T_STEPS / 4)           // 25 float4 per pixel
#define BLOCK_PIXELS     128                     // threads per block (4 waves)
#define ROW_DW           108                     // padded LDS row stride (dwords)
#define CHUNKS_PER_THREAD CHUNKS_PER_PIXEL       // 3200 chunks / 128 threads = 25

typedef __attribute__((ext_vector_type(4))) float f32x4;
typedef __attribute__((ext_vector_type(4))) int   v4i;
typedef __attribute__((address_space(1))) v4i     as1_v4i;   // global int4
typedef __attribute__((address_space(3))) v4i     as3_v4i;   // LDS int4

#if defined(__AMDGCN__) && \
    __has_builtin(__builtin_amdgcn_global_load_async_to_lds_b128) && \
    __has_builtin(__builtin_amdgcn_global_store_async_from_lds_b128)
#define USE_ASYNC_LDS 1
#else
#define USE_ASYNC_LDS 0
#endif

__device__ __forceinline__ void wait_asynccnt0() {
#if __has_builtin(__builtin_amdgcn_s_wait_asynccnt)
  __builtin_amdgcn_s_wait_asynccnt(0);
#else
  asm volatile("s_wait_asynccnt 0x0" ::: "memory");
#endif
}

__global__ __launch_bounds__(BLOCK_PIXELS)
void spike_scan_kernel(const float* __restrict__ u, float* __restrict__ out,
                       int n_pixels) {
  __shared__ float lds[BLOCK_PIXELS * ROW_DW];   // 55,296 B

  const int       tid  = threadIdx.x;
  const long long p0   = (long long)blockIdx.x * BLOCK_PIXELS;

  if (p0 + BLOCK_PIXELS <= (long long)n_pixels) {
    // ---------------- full-tile fast path (block-uniform) ----------------
    const float* gin  = u   + p0 * T_STEPS;
    float*       gout = out + p0 * T_STEPS;

    // Stage 1: coalesced global -> LDS (async on CDNA5), de-swizzle into
    // padded rows. Chunk g covers pixel g/25, time-quad g%25.
#pragma unroll
    for (int i = 0; i < CHUNKS_PER_THREAD; ++i) {
      const int g     = i * BLOCK_PIXELS + tid;
      const int pixel = g / CHUNKS_PER_PIXEL;
      const int k     = g - pixel * CHUNKS_PER_PIXEL;
      const float* gsrc = gin + (long long)g * 4;
      float*       ldst = &lds[pixel * ROW_DW + k * 4];
#if USE_ASYNC_LDS
      __builtin_amdgcn_global_load_async_to_lds_b128(
          (as1_v4i*)(uintptr_t)gsrc,
          (as3_v4i*)(uint32_t)(uintptr_t)ldst,
          /*offset=*/0, /*cpol=*/0);
#else
      *(f32x4*)ldst = *(const f32x4*)gsrc;
#endif
    }
#if USE_ASYNC_LDS
    wait_asynccnt0();
#endif
    __syncthreads();

    // Stage 2: serial scan over this thread's row, overwrite u with spikes.
    {
      float* row = &lds[tid * ROW_DW];
      float  c   = 0.0f;
#pragma unroll
      for (int k = 0; k < CHUNKS_PER_PIXEL; ++k) {
        f32x4 v = *(f32x4*)&row[k * 4];
        f32x4 s;
#pragma unroll
        for (int e = 0; e < 4; ++e) {
          const float sp = (v[e] - c >= 1.0f) ? 1.0f : 0.0f;  // THETA = 1.0
          c += sp;
          s[e] = sp;
        }
        *(f32x4*)&row[k * 4] = s;
      }
    }
    __syncthreads();

    // Stage 3: coalesced LDS -> global (async on CDNA5).
#pragma unroll
    for (int i = 0; i < CHUNKS_PER_THREAD; ++i) {
      const int g     = i * BLOCK_PIXELS + tid;
      const int pixel = g / CHUNKS_PER_PIXEL;
      const int k     = g - pixel * CHUNKS_PER_PIXEL;
      float* gdst = gout + (long long)g * 4;
      float* lsrc = &lds[pixel * ROW_DW + k * 4];
#if USE_ASYNC_LDS
      __builtin_amdgcn_global_store_async_from_lds_b128(
          (as1_v4i*)(uintptr_t)gdst,
          (as3_v4i*)(uint32_t)(uintptr_t)lsrc,
          /*offset=*/0, /*cpol=*/0);
#else
      *(f32x4*)gdst = *(const f32x4*)lsrc;
#endif
    }
#if USE_ASYNC_LDS
    wait_asynccnt0();   // drain before LDS teardown / wave end
#endif
  } else {
    // ---------------- tail path (partial tile; rare/never for this shape) --
    const long long p = p0 + tid;
    if (p < (long long)n_pixels) {
      const float* src = u   + p * T_STEPS;
      float*       dst = out + p * T_STEPS;
      float c = 0.0f;
      for (int t = 0; t < T_STEPS; ++t) {
        const float sp = (src[t] - c >= 1.0f) ? 1.0f : 0.0f;
        c += sp;
        dst[t] = sp;
      }
    }
  }
}

extern "C" void kernel_launch(void* const* d_in, const int* in_sizes, int n_in,
                              void* d_out, int out_size, void* d_ws, size_t ws_size,
                              hipStream_t stream) {
  (void)n_in; (void)out_size; (void)d_ws; (void)ws_size;
  const float* u   = (const float*)d_in[0];
  float*       out = (float*)d_out;

  const long long total    = (long long)in_sizes[0];       // 104,857,600
  const int       n_pixels = (int)(total / T_STEPS);       // 1,048,576
  const int       blocks   = (n_pixels + BLOCK_PIXELS - 1) / BLOCK_PIXELS;  // 8192

  spike_scan_kernel<<<blocks, BLOCK_PIXELS, 0, stream>>>(u, out, n_pixels);
}